// CALayer_20332375179387
// MI455X (gfx1250) — compile-verified
//
#include <hip/hip_runtime.h>
#include <math.h>

typedef __attribute__((ext_vector_type(2))) float v2f;
typedef __attribute__((ext_vector_type(8))) float v8f;

#define GN_EPS 1e-5f
#define N_  32
#define C_  512
#define CR_ 64
#define HW_ 3136   // 56*56

// ---------------------------------------------------------------------------
// Kernel A: global average pool. One 256-thread block per (n,c) row.
// Reads 205 MB -> bandwidth bound, float4 coalesced.
// ---------------------------------------------------------------------------
__global__ void __launch_bounds__(256) pool_kernel(const float* __restrict__ x,
                                                   float* __restrict__ pooled) {
    __shared__ float sdata[256];
    const int row = blockIdx.x;                         // n*C + c
    const float4* xr = (const float4*)(x + (size_t)row * HW_);
    const int nvec = HW_ >> 2;                          // 784
    float s = 0.f;
    for (int i = threadIdx.x; i < nvec; i += 256) {
        float4 v = xr[i];
        s += (v.x + v.y) + (v.z + v.w);
    }
    sdata[threadIdx.x] = s;
    __syncthreads();
    for (int off = 128; off > 0; off >>= 1) {
        if (threadIdx.x < off) sdata[threadIdx.x] += sdata[threadIdx.x + off];
        __syncthreads();
    }
    if (threadIdx.x == 0) pooled[row] = sdata[0] * (1.0f / (float)HW_);
}

// ---------------------------------------------------------------------------
// Kernel B: the whole bottleneck MLP in ONE 256-thread block (8 wave32).
//   h = pooled[32,512] @ w1[64,512]^T + b1      (WMMA f32 16x16x4, 8 tiles)
//   h = ELU(GN(1,64)(h))                        (in LDS)
//   g = h[32,64] @ w2[512,64]^T + b2            (WMMA f32 16x16x4, 64 tiles)
// A-frag (16x4 f32): lanes 0-15 K={k0,k0+1}, lanes 16-31 K={k0+2,k0+3}, M=lane%16.
// B-frag mirrors with N=lane%16. C/D: VGPR v -> M = v + 8*(lane>=16), N=lane%16.
// ---------------------------------------------------------------------------
__global__ void __launch_bounds__(256) mlp_kernel(
    const float* __restrict__ pooled,   // [32,512]
    const float* __restrict__ w1,       // [64,512]
    const float* __restrict__ b1,       // [64]
    const float* __restrict__ gn1w, const float* __restrict__ gn1b,
    const float* __restrict__ w2,       // [512,64]
    const float* __restrict__ b2,       // [512]
    float* __restrict__ g)              // [32,512] out (pre-GN2)
{
    __shared__ float hbuf[N_ * CR_];    // 8 KB
    __shared__ float mu[N_], rs[N_];

    const int tid  = threadIdx.x;
    const int wv   = tid >> 5;          // wave 0..7
    const int lane = tid & 31;
    const int half = lane >> 4;         // 0: lanes 0-15, 1: lanes 16-31
    const int l    = lane & 15;

    // ---- Stage 1: h = pooled @ w1^T  (M=32,N=64,K=512); tile (mt,nt) per wave
    {
        const int mt = wv >> 2, nt = wv & 3;
        const int m = mt * 16 + l;
        const int n = nt * 16 + l;
        v8f acc = {};
        for (int k0 = 0; k0 < C_; k0 += 4) {
            const int kk = k0 + half * 2;
            v2f a, b;
            a.x = pooled[m * C_ + kk];  a.y = pooled[m * C_ + kk + 1];
            b.x = w1[n * C_ + kk];      b.y = w1[n * C_ + kk + 1];
            acc = __builtin_amdgcn_wmma_f32_16x16x4_f32(
                false, a, false, b, (short)0, acc, false, false);
        }
#pragma unroll
        for (int v = 0; v < 8; ++v) {
            const int row = mt * 16 + v + half * 8;
            const int col = nt * 16 + l;
            hbuf[row * CR_ + col] = acc[v] + b1[col];
        }
    }
    __syncthreads();

    // ---- Stage 2: GroupNorm(1, 64) + ELU, per sample (tiny: serial stats ok)
    if (tid < N_) {
        float s = 0.f;
        for (int c = 0; c < CR_; ++c) s += hbuf[tid * CR_ + c];
        const float m_ = s * (1.0f / CR_);
        float s2 = 0.f;
        for (int c = 0; c < CR_; ++c) {
            const float d = hbuf[tid * CR_ + c] - m_;
            s2 += d * d;
        }
        mu[tid] = m_;
        rs[tid] = rsqrtf(s2 * (1.0f / CR_) + GN_EPS);
    }
    __syncthreads();
    for (int idx = tid; idx < N_ * CR_; idx += 256) {
        const int s = idx >> 6, c = idx & (CR_ - 1);
        float val = (hbuf[idx] - mu[s]) * rs[s] * gn1w[c] + gn1b[c];
        hbuf[idx] = (val > 0.f) ? val : expm1f(val);   // ELU alpha=1
    }
    __syncthreads();

    // ---- Stage 3: g = h @ w2^T  (M=32,N=512,K=64); 8 tiles per wave
    for (int mt = 0; mt < 2; ++mt) {
        for (int j = 0; j < 4; ++j) {
            const int nt = wv + j * 8;          // 0..31
            const int m = mt * 16 + l;
            const int n = nt * 16 + l;
            v8f acc = {};
            for (int k0 = 0; k0 < CR_; k0 += 4) {
                const int kk = k0 + half * 2;
                v2f a, b;
                a.x = hbuf[m * CR_ + kk];  a.y = hbuf[m * CR_ + kk + 1];
                b.x = w2[n * CR_ + kk];    b.y = w2[n * CR_ + kk + 1];
                acc = __builtin_amdgcn_wmma_f32_16x16x4_f32(
                    false, a, false, b, (short)0, acc, false, false);
            }
#pragma unroll
            for (int v = 0; v < 8; ++v) {
                const int row = mt * 16 + v + half * 8;
                g[row * C_ + n] = acc[v] + b2[n];
            }
        }
    }
}

// ---------------------------------------------------------------------------
// Kernel C: GroupNorm(1, 512) + sigmoid -> gate. One block per sample.
// ---------------------------------------------------------------------------
__global__ void __launch_bounds__(256) gn2_sigmoid_kernel(
    const float* __restrict__ g,
    const float* __restrict__ gn2w, const float* __restrict__ gn2b,
    float* __restrict__ gate)
{
    __shared__ float ssum[256], ssum2[256];
    const int n = blockIdx.x;
    const float* gr = g + (size_t)n * C_;
    const int c0 = threadIdx.x, c1 = threadIdx.x + 256;
    const float a = gr[c0], b = gr[c1];
    ssum[threadIdx.x]  = a + b;
    ssum2[threadIdx.x] = a * a + b * b;
    __syncthreads();
    for (int off = 128; off > 0; off >>= 1) {
        if (threadIdx.x < off) {
            ssum[threadIdx.x]  += ssum[threadIdx.x + off];
            ssum2[threadIdx.x] += ssum2[threadIdx.x + off];
        }
        __syncthreads();
    }
    const float mean = ssum[0] * (1.0f / C_);
    const float var  = ssum2[0] * (1.0f / C_) - mean * mean;
    const float r = rsqrtf(var + GN_EPS);
    const float v0 = (a - mean) * r * gn2w[c0] + gn2b[c0];
    const float v1 = (b - mean) * r * gn2w[c1] + gn2b[c1];
    gate[(size_t)n * C_ + c0] = 1.0f / (1.0f + expf(-v0));
    gate[(size_t)n * C_ + c1] = 1.0f / (1.0f + expf(-v1));
}

// ---------------------------------------------------------------------------
// Kernel D: out = x * gate[n,c]. One block per (n,c) row, float4 streaming.
// Dominant cost of the whole layer (read 205 MB + write 205 MB).
// ---------------------------------------------------------------------------
__global__ void __launch_bounds__(256) scale_kernel(const float* __restrict__ x,
                                                    const float* __restrict__ gate,
                                                    float* __restrict__ out) {
    const int row = blockIdx.x;                 // n*C + c
    const float gg = gate[row];
    const float4* xr = (const float4*)(x + (size_t)row * HW_);
    float4* orow = (float4*)(out + (size_t)row * HW_);
    const int nvec = HW_ >> 2;                  // 784
    for (int i = threadIdx.x; i < nvec; i += 256) {
        float4 v = xr[i];
        v.x *= gg; v.y *= gg; v.z *= gg; v.w *= gg;
        orow[i] = v;
    }
}

// ---------------------------------------------------------------------------
extern "C" void kernel_launch(void* const* d_in, const int* in_sizes, int n_in,
                              void* d_out, int out_size, void* d_ws, size_t ws_size,
                              hipStream_t stream) {
    const float* x    = (const float*)d_in[0];   // [32,512,56,56]
    const float* w1   = (const float*)d_in[1];   // [64,512]
    const float* b1   = (const float*)d_in[2];   // [64]
    const float* gn1w = (const float*)d_in[3];   // [64]
    const float* gn1b = (const float*)d_in[4];   // [64]
    const float* w2   = (const float*)d_in[5];   // [512,64]
    const float* b2   = (const float*)d_in[6];   // [512]
    const float* gn2w = (const float*)d_in[7];   // [512]
    const float* gn2b = (const float*)d_in[8];   // [512]
    float* out = (float*)d_out;

    float* ws     = (float*)d_ws;
    float* pooled = ws;                    // [32*512]
    float* g      = ws + N_ * C_;          // [32*512]
    float* gate   = ws + 2 * N_ * C_;      // [32*512]

    pool_kernel<<<N_ * C_, 256, 0, stream>>>(x, pooled);
    mlp_kernel<<<1, 256, 0, stream>>>(pooled, w1, b1, gn1w, gn1b, w2, b2, g);
    gn2_sigmoid_kernel<<<N_, 256, 0, stream>>>(g, gn2w, gn2b, gate);
    scale_kernel<<<N_ * C_, 256, 0, stream>>>(x, gate, out);
}